// MultiHeadAttentionLayer_67267777789977
// MI455X (gfx1250) — compile-verified
//
#include <hip/hip_runtime.h>

// MI455X / gfx1250: wave32, WMMA (not MFMA). All GEMMs via v_wmma_f32_16x16x4_f32.
typedef __attribute__((ext_vector_type(2))) float v2f;
typedef __attribute__((ext_vector_type(8))) float v8f;

#define B_    2048
#define N_    32
#define IND   256
#define OUTD  64
#define H_    8
#define QCOLS 512      // OUTD*H

// f32 16x16x4 WMMA per-lane layouts (wave32):
//  A (16x4):  a[r] = A[m][k],  m = lane&15,       k = (lane>>4)*2 + r   (r=0..1)
//  B (4x16):  b[r] = B[k][n],  n = lane&15,       k = (lane>>4)*2 + r
//  C/D(16x16):d[r] = D[m][n],  n = lane&15,       m = r + (lane>>4)*8   (r=0..7)

// ---------------- prep: head-summed V weights (out = attn^T @ (V summed over heads)) ---
__global__ void prep_wvs(const float* __restrict__ WV, const float* __restrict__ bV,
                         float* __restrict__ WVs, float* __restrict__ bVs) {
  int idx = blockIdx.x * blockDim.x + threadIdx.x;   // 0..16383
  int c = idx >> 6;                                  // 0..255
  int j = idx & 63;
  float s = 0.f;
#pragma unroll
  for (int h = 0; h < H_; ++h) s += WV[c * QCOLS + j * H_ + h];
  WVs[c * OUTD + j] = s;
  if (idx < OUTD) {
    float sb = 0.f;
#pragma unroll
    for (int h = 0; h < H_; ++h) sb += bV[idx * H_ + h];
    bVs[idx] = sb;
  }
}

// ---------------- Qh = center @ WQ + bQ : 2048x512x256 ------------------------------
// Each wave owns 4 m-tiles x 1 n-tile: one B fetch feeds 4 WMMAs (4x less WQ traffic).
__global__ __launch_bounds__(256)
void proj_q(const float* __restrict__ center, const float* __restrict__ WQ,
            const float* __restrict__ bQ, float* __restrict__ Qout) {
  int wid  = (blockIdx.x * blockDim.x + threadIdx.x) >> 5;  // 0..1023
  int lane = threadIdx.x & 31;
  int l15 = lane & 15, hi = lane >> 4;
  int g  = wid >> 5;                  // 32 m-groups of 4 tiles (64 rows)
  int nt = wid & 31;                  // 32 n-tiles
  int n0 = nt * 16;
  v8f acc[4] = {{}, {}, {}, {}};
  for (int kt = 0; kt < IND / 4; ++kt) {
    int kb = kt * 4 + hi * 2;
    v2f b;
    b.x = WQ[kb * QCOLS + n0 + l15];
    b.y = WQ[(kb + 1) * QCOLS + n0 + l15];
#pragma unroll
    for (int u = 0; u < 4; ++u) {
      v2f a;
      a.x = center[(size_t)(g * 64 + u * 16 + l15) * IND + kb];
      a.y = center[(size_t)(g * 64 + u * 16 + l15) * IND + kb + 1];
      acc[u] = __builtin_amdgcn_wmma_f32_16x16x4_f32(false, a, false, b, (short)0,
                                                     acc[u], false, false);
    }
  }
  float bias = bQ[n0 + l15];
#pragma unroll
  for (int u = 0; u < 4; ++u)
#pragma unroll
    for (int r = 0; r < 8; ++r)
      Qout[(size_t)(g * 64 + u * 16 + r + hi * 8) * QCOLS + n0 + l15] = acc[u][r] + bias;
}

// ---------------- fused per-b: K-proj + Vs-proj + scores + softmax + reduce -----------
__global__ __launch_bounds__(256)
void attn_fused(const float* __restrict__ neighbors, const float* __restrict__ WK,
                const float* __restrict__ bK, const float* __restrict__ WVs,
                const float* __restrict__ bVs, const float* __restrict__ Qh,
                float* __restrict__ out) {
  __shared__ float nb[N_ * 260];      // neighbors[b], padded stride (260%64=4 -> no bank conflicts)
  __shared__ float Kh[N_ * QCOLS];    // K projection, 64 KB
  __shared__ float Vs[N_ * OUTD];     // head-summed V projection
  __shared__ float Qs[OUTD * 9];      // Q/8, padded stride 9 (conflict-free A reads)
  __shared__ float S[OUTD * 65];      // scores for current n, stride 65
  __shared__ float redM[4 * OUTD];
  __shared__ float redN[4 * OUTD];
  __shared__ float redD[4 * OUTD];

  const int b = blockIdx.x;
  const int t = threadIdx.x;
  const int wave = t >> 5, lane = t & 31, l15 = lane & 15, hi = lane >> 4;

  // Stage neighbors[b] (32x256) and scaled Q row into LDS (coalesced).
  const float* nbg = neighbors + (size_t)b * N_ * IND;
  for (int i = t; i < N_ * IND; i += 256)
    nb[(i >> 8) * 260 + (i & 255)] = nbg[i];
  const float* qg = Qh + (size_t)b * QCOLS;
  for (int i = t; i < QCOLS; i += 256) {
    int d = i >> 3, h = i & 7;
    Qs[d * 9 + h] = qg[i] * 0.125f;   // fold 1/sqrt(OUT_DIM) into Q
  }
  __syncthreads();

  // Kh = nb @ WK + bK : 32x512x256. Both m-tiles share each B fetch -> each block
  // reads WK exactly once (minimal L2 traffic), 2 WMMAs per B operand.
  for (int nt4 = 0; nt4 < 4; ++nt4) {
    int ntile = wave + nt4 * 8;       // 0..31
    int n0 = ntile * 16;
    v8f acc0 = {}, acc1 = {};
    for (int kt = 0; kt < IND / 4; ++kt) {
      int kb = kt * 4 + hi * 2;
      v2f bb, a0, a1;
      bb.x = WK[kb * QCOLS + n0 + l15];
      bb.y = WK[(kb + 1) * QCOLS + n0 + l15];
      a0.x = nb[l15 * 260 + kb];
      a0.y = nb[l15 * 260 + kb + 1];
      a1.x = nb[(16 + l15) * 260 + kb];
      a1.y = nb[(16 + l15) * 260 + kb + 1];
      acc0 = __builtin_amdgcn_wmma_f32_16x16x4_f32(false, a0, false, bb, (short)0,
                                                   acc0, false, false);
      acc1 = __builtin_amdgcn_wmma_f32_16x16x4_f32(false, a1, false, bb, (short)0,
                                                   acc1, false, false);
    }
    float bias = bK[n0 + l15];
#pragma unroll
    for (int r = 0; r < 8; ++r) {
      Kh[(r + hi * 8) * QCOLS + n0 + l15]      = acc0[r] + bias;
      Kh[(16 + r + hi * 8) * QCOLS + n0 + l15] = acc1[r] + bias;
    }
  }

  // Vs = nb @ WVs + bVs : 32x64x256 -> 2x4 tiles, one per wave.
  {
    int mt = wave >> 2, ntile = wave & 3;
    int m0 = mt * 16, n0 = ntile * 16;
    v8f acc = {};
    for (int kt = 0; kt < IND / 4; ++kt) {
      int kb = kt * 4 + hi * 2;
      v2f a, bb;
      a.x = nb[(m0 + l15) * 260 + kb];
      a.y = nb[(m0 + l15) * 260 + kb + 1];
      bb.x = WVs[kb * OUTD + n0 + l15];
      bb.y = WVs[(kb + 1) * OUTD + n0 + l15];
      acc = __builtin_amdgcn_wmma_f32_16x16x4_f32(false, a, false, bb, (short)0, acc,
                                                  false, false);
    }
    float bias = bVs[n0 + l15];
#pragma unroll
    for (int r = 0; r < 8; ++r)
      Vs[(m0 + r + hi * 8) * OUTD + n0 + l15] = acc[r] + bias;
  }
  __syncthreads();

  // Each wave owns 2 score tiles: id = wave*2 + tt; it = id>>2 is the same for both,
  // so the Q A-operands are n-invariant -> hoist them out of the neighbor loop.
  const int it = (wave * 2) >> 2;
  const int m0q = it * 16;
  v2f aQ[2];
#pragma unroll
  for (int ks = 0; ks < 2; ++ks) {
    int kb = ks * 4 + hi * 2;
    aQ[ks].x = Qs[(m0q + l15) * 9 + kb];
    aQ[ks].y = Qs[(m0q + l15) * 9 + kb + 1];
  }

  const int j = t & 63;    // output column owned for reductions
  const int q = t >> 6;    // row-quarter (16 rows each)
  float accOut = 0.f;      // valid for t < 64

  for (int n = 0; n < N_; ++n) {
    // S[i][j] = sum_h Qs[i][h] * Kh[n][j*8+h]   (K=8 -> two chained WMMAs)
#pragma unroll
    for (int tt = 0; tt < 2; ++tt) {
      int id = wave * 2 + tt;
      int j0 = (id & 3) * 16;
      v8f acc = {};
#pragma unroll
      for (int ks = 0; ks < 2; ++ks) {
        int kb = ks * 4 + hi * 2;
        v2f bb;
        bb.x = Kh[n * QCOLS + (j0 + l15) * 8 + kb];
        bb.y = Kh[n * QCOLS + (j0 + l15) * 8 + kb + 1];
        acc = __builtin_amdgcn_wmma_f32_16x16x4_f32(false, aQ[ks], false, bb, (short)0,
                                                    acc, false, false);
      }
#pragma unroll
      for (int r = 0; r < 8; ++r)
        S[(m0q + r + hi * 8) * 65 + j0 + l15] = acc[r];
    }
    __syncthreads();

    // Column softmax over i, fused with Vs dot: 4-way split over row quarters.
    float pm = -3.4e38f;
#pragma unroll
    for (int s = 0; s < 16; ++s)
      pm = fmaxf(pm, S[(q * 16 + s) * 65 + j]);
    redM[q * 64 + j] = pm;
    __syncthreads();
    float m = fmaxf(fmaxf(redM[j], redM[64 + j]), fmaxf(redM[128 + j], redM[192 + j]));
    float pn = 0.f, pd = 0.f;
#pragma unroll
    for (int s = 0; s < 16; ++s) {
      float e = __expf(S[(q * 16 + s) * 65 + j] - m);
      pd += e;
      pn += e * Vs[n * OUTD + q * 16 + s];
    }
    redN[q * 64 + j] = pn;
    redD[q * 64 + j] = pd;
    __syncthreads();
    if (t < 64) {
      float num = redN[j] + redN[64 + j] + redN[128 + j] + redN[192 + j];
      float den = redD[j] + redD[64 + j] + redD[128 + j] + redD[192 + j];
      accOut += num / den;
    }
    // redN/redD reads race only against next-n S writes (disjoint arrays) -> safe.
  }
  if (t < 64) out[(size_t)b * OUTD + t] = accOut * (1.f / H_);
}

extern "C" void kernel_launch(void* const* d_in, const int* in_sizes, int n_in,
                              void* d_out, int out_size, void* d_ws, size_t ws_size,
                              hipStream_t stream) {
  const float* center    = (const float*)d_in[0];
  const float* neighbors = (const float*)d_in[1];
  const float* WQ        = (const float*)d_in[2];
  const float* bQ        = (const float*)d_in[3];
  const float* WK        = (const float*)d_in[4];
  const float* bK        = (const float*)d_in[5];
  const float* WV        = (const float*)d_in[6];
  const float* bV        = (const float*)d_in[7];
  float* out = (float*)d_out;

  float* ws   = (float*)d_ws;
  float* wsQ  = ws;                          // 2048*512 f32 = 4 MB
  float* WVs  = ws + (size_t)B_ * QCOLS;     // 256*64 f32
  float* bVs  = WVs + (size_t)IND * OUTD;    // 64 f32

  prep_wvs<<<64, 256, 0, stream>>>(WV, bV, WVs, bVs);
  proj_q  <<<128, 256, 0, stream>>>(center, WQ, bQ, wsQ);
  attn_fused<<<B_, 256, 0, stream>>>(neighbors, WK, bK, WVs, bVs, wsQ, out);
}